// GraphEncoder_7559142441000
// MI455X (gfx1250) — compile-verified
//
#include <hip/hip_runtime.h>
#include <hip/hip_bf16.h>

#define DIM 128

typedef __attribute__((ext_vector_type(2))) float v2f;
typedef __attribute__((ext_vector_type(8))) float v8f;

// ---------------------------------------------------------------------------
// Degree: deg[n] = 1 + sum_{e: dst[e]==n} ew[e];  then dinv = rsqrt(deg)
// ---------------------------------------------------------------------------
__global__ void init_deg_kernel(float* __restrict__ deg, int n) {
    int i = blockIdx.x * blockDim.x + threadIdx.x;
    if (i < n) deg[i] = 1.0f;   // self-loop weight
}

__global__ void scatter_deg_kernel(const long long* __restrict__ dst,
                                   const float* __restrict__ ew,
                                   float* __restrict__ deg, int e) {
    int i = blockIdx.x * blockDim.x + threadIdx.x;
    if (i < e) atomicAdd(&deg[(int)dst[i]], ew[i]);
}

__global__ void finalize_dinv_kernel(float* __restrict__ deg, int n) {
    int i = blockIdx.x * blockDim.x + threadIdx.x;
    if (i < n) deg[i] = rsqrtf(deg[i]);
}

// ---------------------------------------------------------------------------
// T = act(X) @ W  via V_WMMA_F32_16X16X4_F32 (exact fp32 accumulation).
// Block = 256 threads = 8 waves; each wave owns a 16-row strip x all 128 cols
// (8 column tiles, 8 v8f accumulators). W staged once into LDS (64 KB).
//
// f32 16x16x4 VGPR layout (wave32):
//   A (16x4): VGPR0 = K=0 (lanes 0-15) / K=2 (lanes 16-31); VGPR1 = K=1 / K=3
//   B (4x16): VGPR0 = row K=0 striped over lanes 0-15, row K=2 over 16-31;
//             VGPR1 = K=1 / K=3
//   C/D (16x16): VGPR r -> M=r (lanes 0-15), M=r+8 (lanes 16-31), N=lane%16
// ---------------------------------------------------------------------------
template <int DO_RELU>
__global__ __launch_bounds__(256)
void gemm_wmma_kernel(const float* __restrict__ X, const float* __restrict__ W,
                      float* __restrict__ T, int n) {
    __shared__ float sW[DIM * DIM];  // 64 KB of 320 KB/WGP
    const int tid = threadIdx.x;
    {
        const float4* Wv = (const float4*)W;
        float4* sWv = (float4*)sW;
        #pragma unroll
        for (int i = 0; i < (DIM * DIM / 4) / 256; ++i)
            sWv[tid + i * 256] = Wv[tid + i * 256];
    }
    __syncthreads();

    const int wave = tid >> 5;
    const int lane = tid & 31;
    const int half = lane >> 4;   // 0: K even-low pair, 1: K high pair
    const int l    = lane & 15;
    const int row0 = blockIdx.x * 128 + wave * 16;

    int arow = row0 + l;
    if (arow >= n) arow = n - 1;              // clamp tail; masked on store
    const float* __restrict__ xrow = X + (size_t)arow * DIM;

    v8f acc[8];
    #pragma unroll
    for (int ct = 0; ct < 8; ++ct) acc[ct] = (v8f)0.0f;

    for (int k = 0; k < DIM; k += 4) {
        const int ka = k + 2 * half;          // this half-wave's K pair
        v2f a;
        a.x = xrow[ka];
        a.y = xrow[ka + 1];
        if (DO_RELU) {
            a.x = fmaxf(a.x, 0.0f);
            a.y = fmaxf(a.y, 0.0f);
        }
        #pragma unroll
        for (int ct = 0; ct < 8; ++ct) {
            const int c = ct * 16 + l;
            v2f b;
            b.x = sW[ka * DIM + c];
            b.y = sW[(ka + 1) * DIM + c];
            acc[ct] = __builtin_amdgcn_wmma_f32_16x16x4_f32(
                false, a, false, b, (short)0, acc[ct], false, false);
        }
    }

    #pragma unroll
    for (int r = 0; r < 8; ++r) {
        const int row = row0 + r + 8 * half;
        if (row < n) {
            float* __restrict__ trow = T + (size_t)row * DIM;
            #pragma unroll
            for (int ct = 0; ct < 8; ++ct) trow[ct * 16 + l] = acc[ct][r];
        }
    }
}

// ---------------------------------------------------------------------------
// Z[n][d] = T[n][d] * dinv[n]^2 + bias[d]   (self-loop term + bias)
// ---------------------------------------------------------------------------
__global__ void init_z_kernel(const float* __restrict__ T,
                              const float* __restrict__ dinv,
                              const float* __restrict__ bias,
                              float* __restrict__ Z, int total) {
    int idx = blockIdx.x * blockDim.x + threadIdx.x;
    if (idx < total) {
        int row = idx >> 7;
        int col = idx & (DIM - 1);
        float di = dinv[row];
        Z[idx] = T[idx] * di * di + bias[col];
    }
}

// ---------------------------------------------------------------------------
// One wave32 per edge: Z[dst] += T[src] * (dinv[src]*ew*dinv[dst])
// float4 gather per lane (coalesced 512B/wave), 4 fp32 global atomics.
// ---------------------------------------------------------------------------
__global__ __launch_bounds__(256)
void edge_scatter_kernel(const float* __restrict__ T,
                         const long long* __restrict__ src,
                         const long long* __restrict__ dst,
                         const float* __restrict__ ew,
                         const float* __restrict__ dinv,
                         float* __restrict__ Z, int e) {
    int edge = (int)((blockIdx.x * blockDim.x + threadIdx.x) >> 5);
    int lane = threadIdx.x & 31;
    if (edge >= e) return;
    int s = (int)src[edge];
    int d = (int)dst[edge];
    float nrm = dinv[s] * ew[edge] * dinv[d];
    float4 h = ((const float4*)(T + (size_t)s * DIM))[lane];
    float* zp = Z + (size_t)d * DIM + lane * 4;
    atomicAdd(zp + 0, h.x * nrm);
    atomicAdd(zp + 1, h.y * nrm);
    atomicAdd(zp + 2, h.z * nrm);
    atomicAdd(zp + 3, h.w * nrm);
}

// ---------------------------------------------------------------------------
// In-place ReLU + LayerNorm; one wave32 per row, 4 features/lane.
// ---------------------------------------------------------------------------
__global__ __launch_bounds__(256)
void relu_ln_kernel(float* __restrict__ out, const float* __restrict__ gamma,
                    const float* __restrict__ beta, int n) {
    int row  = blockIdx.x * 8 + (threadIdx.x >> 5);
    int lane = threadIdx.x & 31;
    if (row >= n) return;
    float* p = out + (size_t)row * DIM;
    float4 v = ((const float4*)p)[lane];
    v.x = fmaxf(v.x, 0.0f);
    v.y = fmaxf(v.y, 0.0f);
    v.z = fmaxf(v.z, 0.0f);
    v.w = fmaxf(v.w, 0.0f);

    float s = v.x + v.y + v.z + v.w;
    #pragma unroll
    for (int m = 16; m > 0; m >>= 1) s += __shfl_xor(s, m, 32);
    float mu = s * (1.0f / DIM);

    float ax = v.x - mu, ay = v.y - mu, az = v.z - mu, aw = v.w - mu;
    float q = ax * ax + ay * ay + az * az + aw * aw;
    #pragma unroll
    for (int m = 16; m > 0; m >>= 1) q += __shfl_xor(q, m, 32);
    float rstd = rsqrtf(q * (1.0f / DIM) + 1e-5f);

    float4 g = ((const float4*)gamma)[lane];
    float4 b = ((const float4*)beta)[lane];
    v.x = ax * rstd * g.x + b.x;
    v.y = ay * rstd * g.y + b.y;
    v.z = az * rstd * g.z + b.z;
    v.w = aw * rstd * g.w + b.w;
    ((float4*)p)[lane] = v;
}

// ---------------------------------------------------------------------------
// Host orchestration
// inputs: 0:x 1:edge_index(int64,[2,E]) 2:edge_attr 3:W1 4:b1 5:W2 6:b2
//         7:ln_gamma 8:ln_beta
// ---------------------------------------------------------------------------
extern "C" void kernel_launch(void* const* d_in, const int* in_sizes, int n_in,
                              void* d_out, int out_size, void* d_ws, size_t ws_size,
                              hipStream_t stream) {
    const float*     x    = (const float*)d_in[0];
    const long long* ei   = (const long long*)d_in[1];
    const float*     ew   = (const float*)d_in[2];
    const float*     W1   = (const float*)d_in[3];
    const float*     b1   = (const float*)d_in[4];
    const float*     W2   = (const float*)d_in[5];
    const float*     b2   = (const float*)d_in[6];
    const float*     gam  = (const float*)d_in[7];
    const float*     bet  = (const float*)d_in[8];
    float*           out  = (float*)d_out;

    const int N = in_sizes[0] / DIM;
    const int E = in_sizes[2];
    const long long* src = ei;
    const long long* dst = ei + E;

    // workspace: dinv[N] (padded) | t[N*D] | z1[N*D]
    float* dinv = (float*)d_ws;
    float* t    = dinv + ((N + 1023) & ~1023);
    float* z1   = t + (size_t)N * DIM;

    const int total   = N * DIM;
    const int gNode   = (N + 255) / 256;
    const int gEdgeTh = (E + 255) / 256;
    const int gGemm   = (N + 127) / 128;
    const int gElem   = (total + 255) / 256;
    const int gEdgeWv = (E + 7) / 8;     // 8 waves (edges) per 256-thread block
    const int gLN     = (N + 7) / 8;

    // degrees -> dinv
    init_deg_kernel<<<gNode, 256, 0, stream>>>(dinv, N);
    scatter_deg_kernel<<<gEdgeTh, 256, 0, stream>>>(dst, ew, dinv, E);
    finalize_dinv_kernel<<<gNode, 256, 0, stream>>>(dinv, N);

    // layer 1: t = x @ W1 ; z1 = t*dinv^2 + b1 ; z1 += scatter(t)
    gemm_wmma_kernel<0><<<gGemm, 256, 0, stream>>>(x, W1, t, N);
    init_z_kernel<<<gElem, 256, 0, stream>>>(t, dinv, b1, z1, total);
    edge_scatter_kernel<<<gEdgeWv, 256, 0, stream>>>(t, src, dst, ew, dinv, z1, E);

    // layer 2: t = relu(z1) @ W2 ; out = t*dinv^2 + b2 ; out += scatter(t)
    gemm_wmma_kernel<1><<<gGemm, 256, 0, stream>>>(z1, W2, t, N);
    init_z_kernel<<<gElem, 256, 0, stream>>>(t, dinv, b2, out, total);
    edge_scatter_kernel<<<gEdgeWv, 256, 0, stream>>>(t, src, dst, ew, dinv, out, E);

    // relu + layernorm in place on out
    relu_ln_kernel<<<gLN, 256, 0, stream>>>(out, gam, bet, N);
}